// TDQRLoss_29892972380821
// MI455X (gfx1250) — compile-verified
//
#include <hip/hip_runtime.h>
#include <stdint.h>

// Problem constants (match reference): T=128, B=512, Q=50
#define T_DIM 128
#define B_DIM 512
#define Q_DIM 50
#define GAMMA_F 0.99f
#define TPB 128   // threads per block in loss kernel; block owns TPB (t,b) items

// ---------------------------------------------------------------------------
// CDNA5 async global->LDS copy (ASYNCcnt path).  VDST = LDS byte offset VGPR,
// VADDR = 64-bit global address VGPR pair.
// ---------------------------------------------------------------------------
__device__ __forceinline__ void async_load_b128(uint32_t lds_byte_off, const void* gaddr) {
    asm volatile("global_load_async_to_lds_b128 %0, %1, off"
                 :: "v"(lds_byte_off), "v"(gaddr)
                 : "memory");
}

__device__ __forceinline__ void wait_async0() {
#if __has_builtin(__builtin_amdgcn_s_wait_asynccnt)
    __builtin_amdgcn_s_wait_asynccnt(0);
#else
    asm volatile("s_wait_asynccnt 0" ::: "memory");
#endif
}

// Generic pointer to __shared__ carries the LDS byte offset in its low 32 bits.
__device__ __forceinline__ uint32_t lds_off_of(const void* p) {
    return (uint32_t)(uintptr_t)p;
}

// ---------------------------------------------------------------------------
// Kernel A: quantile-regression Huber loss, 50x50 all-pairs per (t,b).
// Block of TPB threads owns TPB consecutive tb = t*B+b items.
// Rows are staged coalesced into LDS with CDNA5 async b128 loads (ASYNCcnt),
// then each thread runs a pure-VALU 2500-pair loop on its own row pair.
// LDS row stride of 50 dwords is bank-conflict-free for 32 lanes
// (l*50 mod 64 is a bijection onto 32 distinct banks since 25 is odd).
// ---------------------------------------------------------------------------
__global__ __launch_bounds__(TPB) void tdqr_loss_kernel(
    const float* __restrict__ value,    // [T,B,Q]; rows [0, (T-1)*B) used
    const float* __restrict__ returns,  // [T-1,B,Q]
    float*       __restrict__ out)      // [T,B]
{
    __shared__ __align__(16) float sret[TPB * Q_DIM];  // 25.6 KB
    __shared__ __align__(16) float sval[TPB * Q_DIM];  // 25.6 KB

    const int tid = threadIdx.x;
    const int tb0 = blockIdx.x * TPB;
    const int NTB = (T_DIM - 1) * B_DIM;  // 65024, multiple of TPB

    if (tb0 >= NTB) {                     // last time row: loss == 0
        out[tb0 + tid] = 0.0f;
        return;
    }

    // ---- async staging: TPB*Q dwords per array, b128 chunks ---------------
    // global base byte offset = tb0*Q*4 = blk*TPB*200 -> 16B aligned.
    const char* gret = (const char*)(returns + (size_t)tb0 * Q_DIM);
    const char* gval = (const char*)(value   + (size_t)tb0 * Q_DIM);
    const uint32_t lret = lds_off_of(&sret[0]);
    const uint32_t lval = lds_off_of(&sval[0]);
    const int nChunks = (TPB * Q_DIM) / 4;  // 1600 b128 chunks per array
    for (int c = tid; c < nChunks; c += TPB) {
        const uint32_t byteoff = (uint32_t)c * 16u;
        async_load_b128(lret + byteoff, gret + byteoff);
        async_load_b128(lval + byteoff, gval + byteoff);
    }
    wait_async0();       // my wave's async transfers done
    __syncthreads();     // all waves' transfers visible

    // ---- my value row into registers (conflict-free LDS reads) ------------
    const float* vrow = &sval[tid * Q_DIM];
    const float* rrow = &sret[tid * Q_DIM];
    float v[Q_DIM];
#pragma unroll
    for (int j = 0; j < Q_DIM; ++j) v[j] = vrow[j];

    // ---- 50x50 pair loop (VALU-bound core) --------------------------------
    // huber(d) = m*(|d| - 0.5*m), m = min(|d|,1)   (delta = 1)
    // weight   = d<0 ? (1-tau_j) : tau_j
    // 7 VALU ops/pair: sub, min(|.|), fma, mul, cmp, cndmask, fma
    float a0 = 0.0f, a1 = 0.0f, a2 = 0.0f, a3 = 0.0f;
#pragma unroll 2
    for (int i = 0; i < Q_DIM; ++i) {
        const float rv = rrow[i];
#pragma unroll
        for (int j = 0; j < Q_DIM; ++j) {
            const float tau = ((float)j + 0.5f) * (1.0f / (float)Q_DIM);
            const float d  = rv - v[j];
            const float ad = fabsf(d);
            const float m  = fminf(ad, 1.0f);
            const float h  = m * fmaf(-0.5f, m, ad);
            const float w  = (d < 0.0f) ? (1.0f - tau) : tau;
            const float t  = w * h;
            if ((j & 3) == 0)      a0 += t;
            else if ((j & 3) == 1) a1 += t;
            else if ((j & 3) == 2) a2 += t;
            else                   a3 += t;
        }
    }
    const float loss = ((a0 + a1) + (a2 + a3)) * (1.0f / (float)Q_DIM);
    out[tb0 + tid] = loss;
}

// ---------------------------------------------------------------------------
// Kernel B: backward-in-time discounted return scan (branchless).
// One thread per (b,q); loop t = T-2 .. 0.
// returns[t,b,q] = is_last[t,b] ? tv[t,b,q] : (carry*disc[t+1,b]*GAMMA + r[t+1,b])
// ---------------------------------------------------------------------------
__global__ __launch_bounds__(256) void tdqr_scan_kernel(
    const float* __restrict__ reward,        // [T,B]
    const int*   __restrict__ step_type,     // [T,B]
    const float* __restrict__ discount,      // [T,B]
    const float* __restrict__ target_value,  // [T,B,Q]
    float*       __restrict__ returns)       // [T-1,B,Q] (workspace)
{
    const int n = blockIdx.x * blockDim.x + threadIdx.x;   // 0 .. B*Q-1
    if (n >= B_DIM * Q_DIM) return;
    const int b = n / Q_DIM;
    const int BQ = B_DIM * Q_DIM;

    float acc = target_value[(size_t)(T_DIM - 1) * BQ + n];
    for (int t = T_DIM - 2; t >= 0; --t) {
        const float d  = discount[(t + 1) * B_DIM + b] * GAMMA_F;
        const float r  = reward[(t + 1) * B_DIM + b];
        const float tv = target_value[(size_t)t * BQ + n];   // unconditional
        const bool  il = (step_type[t * B_DIM + b] == 2);
        acc = il ? tv : fmaf(acc, d, r);                      // cndmask, no branch
        returns[(size_t)t * BQ + n] = acc;
    }
}

// ---------------------------------------------------------------------------
// Launch wrapper.
// Inputs (setup_inputs order): reward[T,B] f32, step_type[T,B] i32,
// discount[T,B] f32, value[T,B,Q] f32, target_value[T,B,Q] f32.
// Output: loss[T,B] f32.  Workspace: returns[(T-1),B,Q] f32 (~13 MB).
// ---------------------------------------------------------------------------
extern "C" void kernel_launch(void* const* d_in, const int* in_sizes, int n_in,
                              void* d_out, int out_size, void* d_ws, size_t ws_size,
                              hipStream_t stream) {
    const float* reward       = (const float*)d_in[0];
    const int*   step_type    = (const int*)  d_in[1];
    const float* discount     = (const float*)d_in[2];
    const float* value        = (const float*)d_in[3];
    const float* target_value = (const float*)d_in[4];
    float*       out          = (float*)d_out;
    float*       returns      = (float*)d_ws;   // (T-1)*B*Q floats

    (void)in_sizes; (void)n_in; (void)out_size; (void)ws_size;

    // Phase 1: backward scan (parallel over B*Q = 25600 threads).
    const int scanThreads = 256;
    const int scanBlocks  = (B_DIM * Q_DIM + scanThreads - 1) / scanThreads;  // 100
    tdqr_scan_kernel<<<scanBlocks, scanThreads, 0, stream>>>(
        reward, step_type, discount, target_value, returns);

    // Phase 2: quantile loss over all T*B items (last row writes zeros).
    const int lossBlocks = (T_DIM * B_DIM) / TPB;  // 512
    tdqr_loss_kernel<<<lossBlocks, TPB, 0, stream>>>(value, returns, out);
}